// RelativeToAbsoluteModel_4398046511365
// MI455X (gfx1250) — compile-verified
//
#include <hip/hip_runtime.h>

// MI455X / gfx1250, wave32. 3-pass prefix scan; intra-wave 256-element scan done
// with chained v_wmma_f32_16x16x4_f32 (C = L_inc * X, f32-exact).

typedef __attribute__((ext_vector_type(2))) float v2f;
typedef __attribute__((ext_vector_type(8))) float v8f;

// ws layout (floats): [0..4] = per-category step values c1..c5, [8..8+nWaves) = partials
#define WS_PART_OFF 8

__device__ __forceinline__ float stepv(int a, float c1, float c2, float c3,
                                       float c4, float c5) {
  // categories 1..5 -> c1..c5 (4 v_cndmask)
  float r = c5;
  r = (a == 4) ? c4 : r;
  r = (a == 3) ? c3 : r;
  r = (a == 2) ? c2 : r;
  r = (a == 1) ? c1 : r;
  return r;
}

// --- Pass 0: scalar prep (5 clamp values) + out[0] = origin -----------------
__global__ void k0_prep(const float* __restrict__ ow, const float* __restrict__ bdw,
                        const float* __restrict__ dw, const float* __restrict__ sw,
                        const float* __restrict__ iw, const float* __restrict__ biw,
                        float* __restrict__ ws, float* __restrict__ out) {
  if (threadIdx.x == 0) {
    const float EPS = 0.01f;
    float bd = bdw[0], d = dw[0], s = sw[0], in = iw[0], bi = biw[0];
    float c1 = fminf(bd, fminf(0.f, d) - EPS);
    float c2 = fminf(fmaxf(d, bd + EPS), 0.f - EPS);
    float c3 = fminf(fmaxf(s, d + EPS), in - EPS);
    float c4 = fminf(fmaxf(in, 0.f + EPS), bi - EPS);
    float c5 = fmaxf(bi, fmaxf(0.f, in) + EPS);
    ws[0] = c1; ws[1] = c2; ws[2] = c3; ws[3] = c4; ws[4] = c5;
    out[0] = ow[0];
  }
}

// --- Pass 1: per-wave (1024 elem) sums --------------------------------------
__global__ void k1_wave_sums(const int* __restrict__ ann, const float* __restrict__ ws,
                             float* __restrict__ partials, int n, int nWaves) {
  const int lane = threadIdx.x & 31;
  const int wg = blockIdx.x * 8 + (threadIdx.x >> 5);  // 256 thr = 8 waves
  if (wg >= nWaves) return;
  const int base = wg * 1024;
  const float c1 = ws[0], c2 = ws[1], c3 = ws[2], c4 = ws[3], c5 = ws[4];
  float s = 0.f;
  if (base + 1024 <= n) {
    const int4* p = (const int4*)(ann + base);  // 16B aligned
#pragma unroll
    for (int j = 0; j < 8; ++j) {
      int4 a = p[j * 32 + lane];                // fully coalesced b128 loads
      s += stepv(a.x, c1, c2, c3, c4, c5) + stepv(a.y, c1, c2, c3, c4, c5) +
           stepv(a.z, c1, c2, c3, c4, c5) + stepv(a.w, c1, c2, c3, c4, c5);
    }
  } else {
    for (int i = base + lane; i < n; i += 32)
      s += stepv(ann[i], c1, c2, c3, c4, c5);
  }
#pragma unroll
  for (int d = 16; d >= 1; d >>= 1) s += __shfl_xor(s, d, 32);
  if (lane == 0) partials[wg] = s;
}

// --- Pass 2: single-block exclusive scan of partials ------------------------
__global__ void k2_scan_partials(float* __restrict__ partials, int nP) {
  __shared__ float sh[1024];
  const int t = threadIdx.x;
  const int S = (nP + 1023) >> 10;  // per-thread run length (<=64 for n<=2^26)
  float le[64];
  float run = 0.f;
  for (int j = 0; j < S && j < 64; ++j) {
    int idx = t * S + j;
    float v = (idx < nP) ? partials[idx] : 0.f;
    le[j] = run;  // thread-local exclusive
    run += v;
  }
  sh[t] = run;
  __syncthreads();
  for (int d = 1; d < 1024; d <<= 1) {  // Hillis-Steele inclusive
    float v = (t >= d) ? sh[t - d] : 0.f;
    __syncthreads();
    sh[t] += v;
    __syncthreads();
  }
  const float texcl = (t == 0) ? 0.f : sh[t - 1];
  for (int j = 0; j < S && j < 64; ++j) {
    int idx = t * S + j;
    if (idx < nP) partials[idx] = texcl + le[j];
  }
}

// --- Pass 3: WMMA intra-wave scan + emit ------------------------------------
__global__ void k3_scan_emit(const int* __restrict__ ann, const float* __restrict__ ws,
                             const float* __restrict__ partials,
                             const float* __restrict__ ow, float* __restrict__ out,
                             int n, int nWaves) {
  const int lane = threadIdx.x & 31;
  const int wg = blockIdx.x * 8 + (threadIdx.x >> 5);
  if (wg >= nWaves) return;  // wave-uniform: EXEC stays all-1 for WMMA
  const int nIdx = lane & 15;
  const int h = lane >> 4;  // half-wave: carries K+2 / M+8
  const int waveBase = wg * 1024;

  const float c1 = ws[0], c2 = ws[1], c3 = ws[2], c4 = ws[3], c5 = ws[4];
  float carry = ow[0] + partials[wg];

  // Constant A operand: inclusive lower-triangular ones L_inc[m][k] = (k<=m),
  // sliced into four 16x4 K-chunks in the f32 A layout
  // (vgpr v, lane l: M = l&15, K = 4*kc + v + 2*(l>=16)).
  v2f A[4];
#pragma unroll
  for (int kc = 0; kc < 4; ++kc) {
    A[kc].x = ((4 * kc + 0 + 2 * h) <= nIdx) ? 1.f : 0.f;
    A[kc].y = ((4 * kc + 1 + 2 * h) <= nIdx) ? 1.f : 0.f;
  }

  const bool full = (waveBase + 1024 <= n);
  const int lim = (n - 2) & ~1;  // aligned clamp for tail loads (OOB cols unused)

#pragma unroll
  for (int ch = 0; ch < 4; ++ch) {
    const int base = waveBase + ch * 256;
    if (ch < 3) __builtin_prefetch(ann + base + 256 + nIdx * 16, 0, 1);

    // B operand = X column-major: X[k][n] = step[base + n*16 + k]
    // (vgpr v, lane l: K = 4*kc + v + 2*(l>=16), N = l&15) -> int2 pair loads.
    v8f acc = {0.f, 0.f, 0.f, 0.f, 0.f, 0.f, 0.f, 0.f};
#pragma unroll
    for (int kc = 0; kc < 4; ++kc) {
      int idx = base + nIdx * 16 + 4 * kc + 2 * h;
      if (!full) idx = (idx < lim) ? idx : lim;
      int2 av = *(const int2*)(ann + idx);
      v2f b;
      b.x = stepv(av.x, c1, c2, c3, c4, c5);
      b.y = stepv(av.y, c1, c2, c3, c4, c5);
      // acc = L_inc(:,kslice) * X(kslice,:) + acc  -> inclusive column prefix
      acc = __builtin_amdgcn_wmma_f32_16x16x4_f32(false, A[kc], false, b,
                                                  (short)0, acc, false, false);
    }

    // Column totals live in row M=15: C-layout vgpr7, lanes 16..31.
    float ct = __shfl(acc[7], 16 + nIdx, 32);
    // 16-wide inclusive shuffle scan over columns (both halves identical).
    float incl = ct;
#pragma unroll
    for (int d = 1; d < 16; d <<= 1) {
      float u = __shfl_up(incl, d, 16);
      if (nIdx >= d) incl += u;
    }
    const float offs = carry + (incl - ct);  // carry + exclusive col prefix

    // C layout: vgpr r, lane l -> element (M = r + 8h, N = nIdx),
    // linear index = base + nIdx*16 + 8h + r  => 8 consecutive f32 per lane.
    const int outBase = base + nIdx * 16 + 8 * h;
    if (full) {
      float* o = out + 1 + outBase;
#pragma unroll
      for (int r = 0; r < 8; ++r) o[r] = acc[r] + offs;
    } else {
#pragma unroll
      for (int r = 0; r < 8; ++r) {
        int oi = outBase + r;
        if (oi < n) out[1 + oi] = acc[r] + offs;
      }
    }
    carry += __shfl(incl, 31, 32);  // chunk total from lane 31 (N=15)
  }
}

extern "C" void kernel_launch(void* const* d_in, const int* in_sizes, int n_in,
                              void* d_out, int out_size, void* d_ws, size_t ws_size,
                              hipStream_t stream) {
  const int* ann = (const int*)d_in[0];
  const float* ow = (const float*)d_in[1];
  const float* bdw = (const float*)d_in[2];
  const float* dw = (const float*)d_in[3];
  const float* sw = (const float*)d_in[4];
  const float* iw = (const float*)d_in[5];
  const float* biw = (const float*)d_in[6];
  float* out = (float*)d_out;
  float* ws = (float*)d_ws;

  const int n = in_sizes[0];
  const int nWaves = (n + 1023) / 1024;     // 32768 for N=2^25
  float* c = ws;                            // 5 floats
  float* partials = ws + WS_PART_OFF;       // nWaves floats

  k0_prep<<<1, 32, 0, stream>>>(ow, bdw, dw, sw, iw, biw, c, out);
  const int blocks = (nWaves + 7) / 8;      // 8 waves / 256-thread block
  k1_wave_sums<<<blocks, 256, 0, stream>>>(ann, c, partials, n, nWaves);
  k2_scan_partials<<<1, 1024, 0, stream>>>(partials, nWaves);
  k3_scan_emit<<<blocks, 256, 0, stream>>>(ann, c, partials, ow, out, n, nWaves);
}